// Conv2dLocal_29686813950234
// MI455X (gfx1250) — compile-verified
//
#include <hip/hip_runtime.h>
#include <cstdint>

typedef uint32_t u32x4 __attribute__((ext_vector_type(4)));
typedef uint32_t u32x8 __attribute__((ext_vector_type(8)));

// Problem constants (from reference): B=4, C=32, H=W=224, KH=KW=5
#define NB   4
#define NC   32
#define NH   224
#define NW   224
#define NKK  25
#define NL   (NH * NW)      // 50176
#define CKK  (NC * NKK)     // 800
#define ROWK (NW * NKK)     // 5600 dwords staged per block

__global__ __launch_bounds__(NW) void Conv2dLocal_kernel(
    const float* __restrict__ in,   // (B, C, H, W) fp32
    const float* __restrict__ cd,   // (B, L, KK)   fp32
    const float* __restrict__ cwp,  // (1,)
    const float* __restrict__ pwp,  // (1,)
    float* __restrict__ out)        // (B, C, H, W)
{
    __shared__ float s_cd[ROWK];    // 22.4 KB: cd row for this (b, h)
    const int h = blockIdx.x;
    const int b = blockIdx.y;
    const int w = threadIdx.x;      // 224 threads = 7 wave32 waves

    const float* cd_row = cd + ((size_t)b * NL + (size_t)h * NW) * NKK;

    // ---- TDM: one tensor_load_to_lds DMAs the whole 22.4 KB cd row ----
    if (threadIdx.x < 32) {         // wave 0 issues the descriptor (EXEC-independent DMA)
        uint64_t ga = (uint64_t)(uintptr_t)cd_row;                 // 57-bit global byte addr
        uint32_t lds_base = (uint32_t)(uintptr_t)(&s_cd[0]);       // LDS byte offset
        u32x4 g0;
        g0.x = 1u;                                   // count=1, is_restore=0, gather off
        g0.y = lds_base;                             // D#[63:32]  lds_addr
        g0.z = (uint32_t)ga;                         // D#[95:64]  global_addr[31:0]
        g0.w = 0x80000000u | ((uint32_t)(ga >> 32) & 0x01FFFFFFu); // addr[56:32] | type=2
        u32x8 g1;
        g1.s0 = 0x00020000u;                         // workgroup_mask=0, data_size=2 (4B)
        g1.s1 = (ROWK & 0xFFFFu) << 16;              // tensor_dim0[15:0] at bits[63:48]
        g1.s2 = 1u << 16;                            // tensor_dim0[31:16]=0, tensor_dim1=1
        g1.s3 = (uint32_t)ROWK << 16;                // tile_dim0 = 5600
        g1.s4 = 1u;                                  // tile_dim1=1, tile_dim2 unused
        g1.s5 = (uint32_t)ROWK;                      // tensor_dim0_stride[31:0]
        g1.s6 = 0u;                                  // stride0[47:32]=0, stride1 low=0
        g1.s7 = 0u;
        asm volatile("tensor_load_to_lds %0, %1" :: "s"(g0), "s"(g1) : "memory");
        __builtin_amdgcn_s_wait_tensorcnt(0);        // drain TENSORcnt before barrier
    }
    __syncthreads();

    const float cw = cwp[0];
    const float pw = pwp[0];
    const float* inb = in + (size_t)b * NC * NL;
    const int l2 = h * NW + w;

    float acc[NC];
#pragma unroll
    for (int c2 = 0; c2 < NC; ++c2) acc[c2] = 0.0f;

    // For flat unfold index idx = c2*KK*L + k2*L + l2 with 800 = C*KK:
    // the f-decode is c2-independent (c2*KK*L = c2*800*1568) and 1568 = 7*224,
    // so across channels only the source row moves: y += 7, elem offset += 1568.
#pragma unroll 1
    for (int k2 = 0; k2 < NKK; ++k2) {
        int m  = 576 * k2 + l2;          // (k2*L + l2) mod 800 numerator
        int q  = m / CKK;
        int f  = m - q * CKK;
        int c  = f / NKK;                // source channel
        int k  = f - c * NKK;            // 5x5 tap
        int i  = k / 5;
        int j  = k - i * 5;
        int l0 = 62 * k2 + q;            // pixel index at c2 = 0
        int hh = l0 / NW;
        int ww = l0 - hh * NW;
        int x  = ww + j - 2;
        int y0 = hh + i - 2;
        if ((unsigned)x >= (unsigned)NW) y0 = 1 << 20;  // fold x-OOB into y check
        int e  = c * NL + y0 * NW + x;   // element offset (only used when valid)

        // fused weight: pos(k2)*pw + cd*cw  (shared by all 32 channels)
        int dj = k2 - (k2 / 5) * 5 - 2;
        int di = k2 / 5 - 2;
        float wk = (float)(di * di + dj * dj) * pw + s_cd[w * NKK + k2] * cw;

#pragma unroll
        for (int c2 = 0; c2 < NC; ++c2) {
            int y = y0 + 7 * c2;
            bool ok = (unsigned)y < (unsigned)NH;
            int eo = ok ? e : 0;         // safe clamped address
            float v = inb[eo];
            acc[c2] += (ok ? v : 0.0f) * wk;
            e += 7 * NW;                 // += 1568
        }
    }

    size_t o = (size_t)b * NC * NL + (size_t)h * NW + w;
#pragma unroll
    for (int c2 = 0; c2 < NC; ++c2) {
        out[o] = acc[c2] * in[o];
        o += NL;
    }
}

extern "C" void kernel_launch(void* const* d_in, const int* in_sizes, int n_in,
                              void* d_out, int out_size, void* d_ws, size_t ws_size,
                              hipStream_t stream) {
    (void)in_sizes; (void)n_in; (void)out_size; (void)d_ws; (void)ws_size;
    const float* in  = (const float*)d_in[0];
    const float* cd  = (const float*)d_in[1];
    const float* cwp = (const float*)d_in[2];
    const float* pwp = (const float*)d_in[3];
    float* out = (float*)d_out;

    dim3 grid(NH, NB);   // one block per (row, batch): 896 blocks
    Conv2dLocal_kernel<<<grid, NW, 0, stream>>>(in, cd, cwp, pwp, out);
}